// Fusedmax_13211319402876
// MI455X (gfx1250) — compile-verified
//
#include <hip/hip_runtime.h>
#include <stdint.h>

// Fusedmax on MI455X (gfx1250, wave32):
//   phase 0: async global->LDS tile load (ASYNCcnt path, b128 per lane)
//   phase 1: Condat TV1D prox, one thread per row, row resident in LDS
//   phase 2: sparsemax per row via wave32 bisection on sum(max(z-tau,0))=1
//
// B=2048 rows, N=128 cols, fp32. One wave per block, 32 rows per block.

#define TVN      128      // row length
#define RPB      32       // rows per block (== lanes per wave)
#define RSTRIDE  132      // padded row stride in dwords: 528B, 16B aligned,
                          // 132%64==4 -> good LDS bank spread across rows
#define LAM      1.0f     // ALPHA

__global__ __launch_bounds__(32)
void fusedmax_tv_sparsemax(const float* __restrict__ x,
                           float* __restrict__ out,
                           int nrows)
{
    __shared__ float smem[RPB * RSTRIDE];
    const int lane = threadIdx.x;                 // 0..31 (wave32)
    const int row0 = blockIdx.x * RPB;

    // ---------------- Phase 0: async global -> LDS (ASYNCcnt) -------------
    // GVS addressing: addr = SGPR64 + VGPR32 + inst_offset. Each lane moves
    // 16 bytes per issue; one issue covers 512B = one full row per wave.
    {
        const uint64_t gbase = (uint64_t)(uintptr_t)x + (uint64_t)row0 * (TVN * 4u);
        #pragma unroll
        for (int r = 0; r < RPB; ++r) {
            uint32_t ldsaddr = (uint32_t)(uintptr_t)(&smem[r * RSTRIDE])
                             + (uint32_t)(lane * 16);
            uint32_t goff = (uint32_t)(r * (TVN * 4) + lane * 16);
            asm volatile("global_load_async_to_lds_b128 %0, %1, %2"
                         :: "v"(ldsaddr), "v"(goff), "s"(gbase)
                         : "memory");
        }
        asm volatile("s_wait_asynccnt 0" ::: "memory");
    }
    __syncthreads();   // single-wave block: compiles to counter waits only

    // ---------------- Phase 1: Condat TV1D prox (one thread per row) ------
    // Faithful in-place translation of the reference state machine
    // (w,k,k0,kminus,kplus,vmin,vmax,umin,umax). Sequential by nature.
    {
        float* w = &smem[lane * RSTRIDE];
        int k = 0, k0 = 0, km = 0, kp = 0;
        float vmin = w[0] - LAM, vmax = w[0] + LAM;
        float umin = LAM, umax = -LAM;
        bool done = false;
        for (int it = 0; it < 40000 && !done; ++it) {   // O(N^2) hard cap
            if (k == TVN - 1) {
                if (umin < 0.0f) {                       // case A
                    for (int i = k0; i <= km; ++i) w[i] = vmin;
                    k = k0 = km = km + 1;
                    vmin = w[k];
                    umax = vmin + LAM - vmax;
                    umin = LAM;
                } else if (umax > 0.0f) {                // case B
                    for (int i = k0; i <= kp; ++i) w[i] = vmax;
                    k = k0 = kp = kp + 1;
                    vmax = w[k];
                    umin = vmax - LAM - vmin;
                    umax = -LAM;
                } else {                                 // case C: terminate
                    const float v = vmin + umin / (float)(k - k0 + 1);
                    for (int i = k0; i <= k; ++i) w[i] = v;
                    done = true;
                }
            } else {
                const float wk1 = w[k + 1];
                if (wk1 + umin < vmin - LAM) {           // case D
                    for (int i = k0; i <= km; ++i) w[i] = vmin;
                    k = k0 = km = kp = km + 1;
                    vmin = w[k];
                    vmax = vmin + 2.0f * LAM;
                    umin = LAM;  umax = -LAM;
                } else if (wk1 + umax > vmax + LAM) {    // case E
                    for (int i = k0; i <= kp; ++i) w[i] = vmax;
                    k = k0 = km = kp = kp + 1;
                    vmax = w[k];
                    vmin = vmax - 2.0f * LAM;
                    umin = LAM;  umax = -LAM;
                } else {                                 // case F
                    k += 1;
                    umin += wk1 - vmin;
                    umax += wk1 - vmax;
                    if (umin >= LAM) {
                        vmin += (umin - LAM) / (float)(k - k0 + 1);
                        umin = LAM;  km = k;
                    }
                    if (umax <= -LAM) {
                        vmax += (umax + LAM) / (float)(k - k0 + 1);
                        umax = -LAM; kp = k;
                    }
                }
            }
        }
    }
    __syncthreads();

    // ---------------- Phase 2: sparsemax per row (wave32 bisection) -------
    // tau solves sum(max(z - tau, 0)) = 1; s(tau) is monotone decreasing,
    // tau in [max(z) - 1, max(z)]. 32 halvings >> fp32 resolution.
    for (int r = 0; r < RPB; ++r) {
        const int row = row0 + r;
        if (row >= nrows) break;

        const float4 z4 = *(const float4*)&smem[r * RSTRIDE + lane * 4];

        // wave-wide max (xor butterfly, width 32)
        float m = fmaxf(fmaxf(z4.x, z4.y), fmaxf(z4.z, z4.w));
        #pragma unroll
        for (int off = 16; off >= 1; off >>= 1)
            m = fmaxf(m, __shfl_xor(m, off, 32));

        float lo = m - 1.0f, hi = m;
        #pragma unroll 4
        for (int it = 0; it < 32; ++it) {
            const float tau = 0.5f * (lo + hi);
            float s = fmaxf(z4.x - tau, 0.0f) + fmaxf(z4.y - tau, 0.0f)
                    + fmaxf(z4.z - tau, 0.0f) + fmaxf(z4.w - tau, 0.0f);
            #pragma unroll
            for (int off = 16; off >= 1; off >>= 1)
                s += __shfl_xor(s, off, 32);
            if (s >= 1.0f) lo = tau; else hi = tau;     // uniform branch
        }
        const float tau = 0.5f * (lo + hi);

        float4 o;
        o.x = fmaxf(z4.x - tau, 0.0f);
        o.y = fmaxf(z4.y - tau, 0.0f);
        o.z = fmaxf(z4.z - tau, 0.0f);
        o.w = fmaxf(z4.w - tau, 0.0f);
        *(float4*)&out[(size_t)row * TVN + lane * 4] = o;   // b128 store
    }
}

extern "C" void kernel_launch(void* const* d_in, const int* in_sizes, int n_in,
                              void* d_out, int out_size, void* d_ws, size_t ws_size,
                              hipStream_t stream) {
    const float* x = (const float*)d_in[0];
    float* out = (float*)d_out;
    const int nrows = in_sizes[0] / TVN;           // 2048
    const int grid  = (nrows + RPB - 1) / RPB;     // 64 blocks x 1 wave
    fusedmax_tv_sparsemax<<<grid, 32, 0, stream>>>(x, out, nrows);
}